// VQVAE_CNN_43026982372007
// MI455X (gfx1250) — compile-verified
//
#include <hip/hip_runtime.h>
#include <math.h>

typedef float v2f __attribute__((ext_vector_type(2)));
typedef float v8f __attribute__((ext_vector_type(8)));
typedef int   v4i __attribute__((ext_vector_type(4)));

#define N_ROWS 131072
#define T_STEPS 10
#define NT (N_ROWS * T_STEPS) // 1310720

// ---------------- workspace layout (float offsets) ----------------
constexpr size_t WS_H0  = 0;                                  // NT x 48 encoder pre-BN
constexpr size_t WS_C1  = WS_H0 + (size_t)NT * 48;            // N x 32 x 4 conv1 pre-BN
constexpr size_t WS_C2  = WS_C1 + (size_t)N_ROWS * 128;       // N x 16 x 3 conv2 pre-BN
constexpr size_t WS_E0  = WS_C2 + (size_t)N_ROWS * 48;        // N x 64 dec0 pre-BN
constexpr size_t WS_E1  = WS_E0 + (size_t)N_ROWS * 64;        // N x 128 dec1 pre-BN
constexpr size_t WS_P0  = WS_E1 + (size_t)N_ROWS * 128;       // 64 slots x (48 sum + 48 sq)
constexpr size_t WS_P1  = WS_P0 + 64 * 96;
constexpr size_t WS_P2  = WS_P1 + 64 * 64;
constexpr size_t WS_PD0 = WS_P2 + 64 * 32;
constexpr size_t WS_PD1 = WS_PD0 + 64 * 128;
constexpr size_t WS_S0  = WS_PD1 + 64 * 256;                  // scale[F],shift[F] per BN
constexpr size_t WS_S1  = WS_S0 + 96;
constexpr size_t WS_S2  = WS_S1 + 64;
constexpr size_t WS_SD0 = WS_S2 + 32;
constexpr size_t WS_SD1 = WS_SD0 + 128;
constexpr int PART_TOTAL = 64 * (96 + 64 + 32 + 128 + 256);   // 36864 floats

// ---------------- output layout (tuple concat) ----------------
constexpr size_t O_IH = 0;                                    // N x 45
constexpr size_t O_Q  = (size_t)N_ROWS * 45;                  // N x 16
constexpr size_t O_Z  = O_Q + (size_t)N_ROWS * 16;            // N x 16
constexpr size_t O_V  = O_Z + (size_t)N_ROWS * 16;            // N x 3

static __device__ __forceinline__ float eluf(float x) {
  return x > 0.f ? x : (expf(x) - 1.f);
}

// CDNA5 native fp32 WMMA: D(16x16) = A(16x4) * B(4x16) + C
static __device__ __forceinline__ v8f wmma4(v2f a, v2f b, v8f c) {
  return __builtin_amdgcn_wmma_f32_16x16x4_f32(false, a, false, b, (short)0, c,
                                               false, false);
}

// ---- CDNA5 async global->LDS DMA (ASYNCcnt-tracked), guarded fallback ----
#if defined(__HIP_DEVICE_COMPILE__) &&                                         \
    __has_builtin(__builtin_amdgcn_global_load_async_to_lds_b128) &&           \
    __has_builtin(__builtin_amdgcn_s_wait_asynccnt)
#define HAVE_ASYNC_LDS 1
typedef __attribute__((address_space(1))) v4i g_v4i;
typedef __attribute__((address_space(3))) v4i l_v4i;
#else
#define HAVE_ASYNC_LDS 0
#endif

// copy nFloats (multiple of 4, 16B-aligned src+dst) from global to LDS
static __device__ __forceinline__ void lds_cp128(float* dstL, const float* srcG,
                                                 int nFloats, int t, int stride) {
#if HAVE_ASYNC_LDS
  for (int j = t * 4; j < nFloats; j += stride * 4)
    __builtin_amdgcn_global_load_async_to_lds_b128(
        (g_v4i*)(srcG + j), (l_v4i*)(dstL + j), 0, 0);
#else
  for (int j = t; j < nFloats; j += stride) dstL[j] = srcG[j];
#endif
}
static __device__ __forceinline__ void lds_cp_wait() {
#if HAVE_ASYNC_LDS
  __builtin_amdgcn_s_wait_asynccnt(0);
#endif
}

// ---------------- utility kernels ----------------
__global__ void k_zero(float* __restrict__ p, int n) {
  int i = blockIdx.x * blockDim.x + threadIdx.x;
  if (i < n) p[i] = 0.f;
}

// fold 64 partial slots -> per-feature scale/shift
__global__ void k_bnfin(const float* __restrict__ part, const float* __restrict__ g,
                        const float* __restrict__ b, float* __restrict__ ss, int F,
                        float invCnt) {
  int f = threadIdx.x;
  if (f < F) {
    float s = 0.f, q = 0.f;
    for (int k = 0; k < 64; ++k) {
      s += part[k * 2 * F + f];
      q += part[k * 2 * F + F + f];
    }
    float m  = s * invCnt;
    float v  = q * invCnt - m * m;
    float sc = g[f] * rsqrtf(v + 1e-5f);
    ss[f]     = sc;
    ss[F + f] = b[f] - m * sc;
  }
}

// ---------------- K1: encoder GEMM (WMMA f32) + BN0 stats ----------------
__global__ __launch_bounds__(256) void k_enc(const float* __restrict__ obs,
                                             const float* __restrict__ enc_w,
                                             const float* __restrict__ enc_b,
                                             float* __restrict__ h0,
                                             float* __restrict__ part) {
  __shared__ float BtE[48 * 48];      // [out][k], k padded 45->48 (transform -> manual)
  __shared__ float aT[8][16 * 48];    // per-wave A tile (padded -> manual)
  __shared__ float sSt[96];
  int tid = threadIdx.x, lane = tid & 31, wv = tid >> 5;

  for (int j = tid; j < 48 * 48; j += 256) {
    int o = j / 48, k = j % 48;
    BtE[j] = (k < 45) ? enc_w[o * 45 + k] : 0.f;
  }
  if (tid < 96) sSt[tid] = 0.f;

  int rowBase = (blockIdx.x * 8 + wv) * 16;
  for (int j = lane; j < 16 * 48; j += 32) {
    int r = j / 48, c = j % 48;
    aT[wv][j] = (c < 45) ? obs[(size_t)(rowBase + r) * 45 + c] : 0.f;
  }
  __syncthreads();

  int m16 = lane & 15, halfk = (lane >> 4) << 1, rof = (lane >> 4) << 3;
  v8f acc[3];
#pragma unroll
  for (int ct = 0; ct < 3; ++ct) acc[ct] = {0.f, 0.f, 0.f, 0.f, 0.f, 0.f, 0.f, 0.f};
#pragma unroll
  for (int s = 0; s < 12; ++s) {
    int ko = 4 * s + halfk;
    v2f a = *(const v2f*)&aT[wv][m16 * 48 + ko];
#pragma unroll
    for (int ct = 0; ct < 3; ++ct) {
      v2f b = *(const v2f*)&BtE[(ct * 16 + m16) * 48 + ko];
      acc[ct] = wmma4(a, b, acc[ct]);
    }
  }
#pragma unroll
  for (int ct = 0; ct < 3; ++ct) {
    int feat = ct * 16 + m16;
    float bia = enc_b[feat];
    float s_ = 0.f, q_ = 0.f;
#pragma unroll
    for (int i = 0; i < 8; ++i) {
      float v = acc[ct][i] + bia;
      h0[(size_t)(rowBase + i + rof) * 48 + feat] = v;
      s_ += v; q_ += v * v;
    }
    atomicAdd(&sSt[feat], s_);
    atomicAdd(&sSt[48 + feat], q_);
  }
  __syncthreads();
  if (tid < 96) atomicAdd(&part[(blockIdx.x & 63) * 96 + tid], sSt[tid]);
}

// ---------------- K3: BN0+ELU -> conv1 (k=4,s=2) + BN1 stats ----------------
// 8 samples per block (4 iterations x 2 subgroups), weights staged once.
__global__ __launch_bounds__(256) void k_conv1(const float* __restrict__ h0,
                                               const float* __restrict__ ss0,
                                               const float* __restrict__ w1g,
                                               const float* __restrict__ b1,
                                               float* __restrict__ c1,
                                               float* __restrict__ part) {
  __shared__ float w1[32 * 192];                           // [oc][ti][c] (transposed)
  __shared__ __attribute__((aligned(16))) float xs[2][480];
  __shared__ float sSt[64];
  int tid = threadIdx.x;
  for (int j = tid; j < 6144; j += 256) {
    int oc = j / 192, r = j % 192, ti = r / 48, c = r % 48;
    w1[j] = w1g[oc * 192 + c * 4 + ti];
  }
  if (tid < 64) sSt[tid] = 0.f;
  int sub = tid >> 7, local = tid & 127;
  int p = local >> 5, oc = local & 31;
  float sAcc = 0.f, qAcc = 0.f;
  for (int it = 0; it < 4; ++it) {
    int n = blockIdx.x * 8 + it * 2 + sub;
    lds_cp128(xs[sub], h0 + (size_t)n * 480, 480, local, 128);  // raw DMA
    lds_cp_wait();
    __syncthreads();
    for (int j = local; j < 480; j += 128) {                    // BN+ELU in place
      int c = j % 48;
      xs[sub][j] = eluf(fmaf(xs[sub][j], ss0[c], ss0[48 + c]));
    }
    __syncthreads();
    const float* xr = &xs[sub][2 * p * 48];
    const float* wr = &w1[oc * 192];
    float acc = 0.f;
#pragma unroll 8
    for (int j = 0; j < 192; ++j) acc += xr[j] * wr[j];
    float v = acc + b1[oc];
    c1[(size_t)n * 128 + oc * 4 + p] = v;
    sAcc += v; qAcc += v * v;
    __syncthreads();                                            // protect xs reuse
  }
  atomicAdd(&sSt[oc], sAcc);
  atomicAdd(&sSt[32 + oc], qAcc);
  __syncthreads();
  if (tid < 64) atomicAdd(&part[(blockIdx.x & 63) * 64 + tid], sSt[tid]);
}

// ---------------- K5: BN1+ELU -> conv2 (k=2,s=1) + BN2 stats ----------------
// 16 samples per block (4 iterations x 4 subgroups).
__global__ __launch_bounds__(256) void k_conv2(const float* __restrict__ c1,
                                               const float* __restrict__ ss1,
                                               const float* __restrict__ w2g,
                                               const float* __restrict__ b2,
                                               float* __restrict__ c2,
                                               float* __restrict__ part) {
  __shared__ __attribute__((aligned(16))) float w2[16 * 64];  // (16,32,2) as-is
  __shared__ __attribute__((aligned(16))) float xs[4][128];
  __shared__ float sSt[32];
  int tid = threadIdx.x;
  lds_cp128(w2, w2g, 1024, tid, 256);
  if (tid < 32) sSt[tid] = 0.f;
  int sub = tid >> 6, local = tid & 63;
  int oc = local / 3, p = local % 3;
  float sAcc = 0.f, qAcc = 0.f;
  for (int it = 0; it < 4; ++it) {
    int n = blockIdx.x * 16 + it * 4 + sub;
    lds_cp128(xs[sub], c1 + (size_t)n * 128, 128, local, 64);
    lds_cp_wait();
    __syncthreads();
    for (int j = local; j < 128; j += 64) {
      int ic = j >> 2;
      xs[sub][j] = eluf(fmaf(xs[sub][j], ss1[ic], ss1[32 + ic]));
    }
    __syncthreads();
    if (local < 48) {
      float acc = 0.f;
#pragma unroll
      for (int ic = 0; ic < 32; ++ic) {
        acc += xs[sub][ic * 4 + p]     * w2[oc * 64 + ic * 2 + 0];
        acc += xs[sub][ic * 4 + p + 1] * w2[oc * 64 + ic * 2 + 1];
      }
      float v = acc + b2[oc];
      c2[(size_t)n * 48 + oc * 3 + p] = v;
      sAcc += v; qAcc += v * v;
    }
    __syncthreads();
  }
  if (local < 48) {
    atomicAdd(&sSt[oc], sAcc);
    atomicAdd(&sSt[16 + oc], qAcc);
  }
  __syncthreads();
  if (tid < 32) atomicAdd(&part[(blockIdx.x & 63) * 32 + tid], sSt[tid]);
}

// ---------------- K7: BN2+ELU -> z/vel/VQ/q -> e0 (all WMMA) ----------------
__global__ __launch_bounds__(256) void k_head(
    const float* __restrict__ c2, const float* __restrict__ ss2,
    const float* __restrict__ lin_w, const float* __restrict__ lin_b,
    const float* __restrict__ vel_w, const float* __restrict__ vel_b,
    const float* __restrict__ cbG, const float* __restrict__ d0w,
    const float* __restrict__ d0b, float* __restrict__ e0,
    float* __restrict__ part, float* __restrict__ outQ,
    float* __restrict__ outZ, float* __restrict__ outV) {
  __shared__ __attribute__((aligned(16))) float cb[512 * 16];
  __shared__ float cn2[512];
  __shared__ __attribute__((aligned(16))) float BL[16 * 48];  // lin_w [out][k]
  __shared__ __attribute__((aligned(16))) float B0[64 * 16];  // d0_w [out][k]
  __shared__ __attribute__((aligned(16))) float hT[8][16 * 48];
  __shared__ float zT[8][256];
  __shared__ float qT[8][256];
  __shared__ float sSt[128];
  __shared__ int   idxS[8][16];
  int tid = threadIdx.x, lane = tid & 31, wv = tid >> 5;
  int rowBase = (blockIdx.x * 8 + wv) * 16;

  lds_cp128(cb, cbG, 8192, tid, 256);
  lds_cp128(BL, lin_w, 768, tid, 256);
  lds_cp128(B0, d0w, 1024, tid, 256);
  lds_cp128(hT[wv], c2 + (size_t)rowBase * 48, 768, lane, 32);  // raw c2 tile
  if (tid < 128) sSt[tid] = 0.f;
  lds_cp_wait();
  __syncthreads();
  for (int j = tid; j < 512; j += 256) {
    float s = 0.f;
    for (int k = 0; k < 16; ++k) { float c = cb[j * 16 + k]; s += c * c; }
    cn2[j] = s;
  }
  for (int j = lane; j < 768; j += 32) {                        // BN2+ELU in place
    int ch = (j % 48) / 3;
    hT[wv][j] = eluf(fmaf(hT[wv][j], ss2[ch], ss2[16 + ch]));
  }
  __syncthreads();

  int m16 = lane & 15, halfk = (lane >> 4) << 1, rof = (lane >> 4) << 3;
  // z = h @ lin_w^T + b
  v8f zacc = {0.f, 0.f, 0.f, 0.f, 0.f, 0.f, 0.f, 0.f};
#pragma unroll
  for (int s = 0; s < 12; ++s) {
    int ko = 4 * s + halfk;
    v2f a = *(const v2f*)&hT[wv][m16 * 48 + ko];
    v2f b = *(const v2f*)&BL[m16 * 48 + ko];
    zacc = wmma4(a, b, zacc);
  }
  {
    float bia = lin_b[m16];
#pragma unroll
    for (int i = 0; i < 8; ++i) zT[wv][(i + rof) * 16 + m16] = zacc[i] + bia;
  }
  __syncthreads();
  // L2-normalize z rows, write z + vel
  if (lane < 16) {
    int row = lane;
    float s = 0.f;
    for (int k = 0; k < 16; ++k) { float z = zT[wv][row * 16 + k]; s += z * z; }
    float inv = 1.f / fmaxf(sqrtf(s), 1e-12f);
    for (int k = 0; k < 16; ++k) {
      float z = zT[wv][row * 16 + k] * inv;
      zT[wv][row * 16 + k] = z;
      outZ[(size_t)(rowBase + row) * 16 + k] = z;
    }
    for (int j = 0; j < 3; ++j) {
      float acc = 0.f;
      for (int k = 0; k < 48; ++k) acc += hT[wv][row * 48 + k] * vel_w[j * 48 + k];
      outV[(size_t)(rowBase + row) * 3 + j] = acc + vel_b[j];
    }
  }
  __syncthreads();
  // VQ: argmin_j (|c_j|^2 - 2 z.c_j) via WMMA G = z @ cb^T
  float bestv[8];
  int besti[8];
#pragma unroll
  for (int i = 0; i < 8; ++i) { bestv[i] = 3.4e38f; besti[i] = 0; }
  for (int ct = 0; ct < 32; ++ct) {
    v8f g = {0.f, 0.f, 0.f, 0.f, 0.f, 0.f, 0.f, 0.f};
#pragma unroll
    for (int s = 0; s < 4; ++s) {
      int ko = 4 * s + halfk;
      v2f a = *(const v2f*)&zT[wv][m16 * 16 + ko];
      v2f b = *(const v2f*)&cb[(ct * 16 + m16) * 16 + ko];
      g = wmma4(a, b, g);
    }
    int j = ct * 16 + m16;
    float cj = cn2[j];
#pragma unroll
    for (int i = 0; i < 8; ++i) {
      float sc = cj - 2.f * g[i];
      if (sc < bestv[i] || (sc == bestv[i] && j < besti[i])) { bestv[i] = sc; besti[i] = j; }
    }
  }
#pragma unroll
  for (int i = 0; i < 8; ++i) {
#pragma unroll
    for (int m = 8; m >= 1; m >>= 1) {
      float ov = __shfl_xor(bestv[i], m, 32);
      int   oi = __shfl_xor(besti[i], m, 32);
      if (ov < bestv[i] || (ov == bestv[i] && oi < besti[i])) { bestv[i] = ov; besti[i] = oi; }
    }
  }
  if (lane == 0)  for (int i = 0; i < 8; ++i) idxS[wv][i] = besti[i];
  if (lane == 16) for (int i = 0; i < 8; ++i) idxS[wv][8 + i] = besti[i];
  __syncthreads();
  // q = l2norm(codebook[idx])   (straight-through forward value)
  if (lane < 16) {
    int row = lane, id = idxS[wv][row];
    float s = 0.f;
    for (int k = 0; k < 16; ++k) { float c = cb[id * 16 + k]; s += c * c; }
    float inv = 1.f / fmaxf(sqrtf(s), 1e-12f);
    for (int k = 0; k < 16; ++k) {
      float q = cb[id * 16 + k] * inv;
      qT[wv][row * 16 + k] = q;
      outQ[(size_t)(rowBase + row) * 16 + k] = q;
    }
  }
  __syncthreads();
  // e0 = q @ d0_w^T + b (pre-BN) + stats
  v8f acc[4];
#pragma unroll
  for (int ct = 0; ct < 4; ++ct) acc[ct] = {0.f, 0.f, 0.f, 0.f, 0.f, 0.f, 0.f, 0.f};
#pragma unroll
  for (int s = 0; s < 4; ++s) {
    int ko = 4 * s + halfk;
    v2f a = *(const v2f*)&qT[wv][m16 * 16 + ko];
#pragma unroll
    for (int ct = 0; ct < 4; ++ct) {
      v2f b = *(const v2f*)&B0[(ct * 16 + m16) * 16 + ko];
      acc[ct] = wmma4(a, b, acc[ct]);
    }
  }
#pragma unroll
  for (int ct = 0; ct < 4; ++ct) {
    int col = ct * 16 + m16;
    float bia = d0b[col];
    float s_ = 0.f, q_ = 0.f;
#pragma unroll
    for (int i = 0; i < 8; ++i) {
      float v = acc[ct][i] + bia;
      e0[(size_t)(rowBase + i + rof) * 64 + col] = v;
      s_ += v; q_ += v * v;
    }
    atomicAdd(&sSt[col], s_);
    atomicAdd(&sSt[64 + col], q_);
  }
  __syncthreads();
  if (tid < 128) atomicAdd(&part[(blockIdx.x & 63) * 128 + tid], sSt[tid]);
}

// ---------------- K9: elu(bnd0(e0)) @ d1_w^T (WMMA) + BND1 stats ----------------
__global__ __launch_bounds__(256) void k_d1(const float* __restrict__ e0,
                                            const float* __restrict__ ssd0,
                                            const float* __restrict__ d1w,
                                            const float* __restrict__ d1b,
                                            float* __restrict__ e1,
                                            float* __restrict__ part) {
  __shared__ __attribute__((aligned(16))) float B[128 * 64];   // d1_w [out][k]
  __shared__ __attribute__((aligned(16))) float aT[8][16 * 64];
  __shared__ float sSt[256];
  int tid = threadIdx.x, lane = tid & 31, wv = tid >> 5;
  int rowBase = (blockIdx.x * 8 + wv) * 16;
  lds_cp128(B, d1w, 8192, tid, 256);
  lds_cp128(aT[wv], e0 + (size_t)rowBase * 64, 1024, lane, 32);
  sSt[tid] = 0.f;
  lds_cp_wait();
  __syncthreads();
  for (int j = lane; j < 1024; j += 32) {
    int f = j & 63;
    aT[wv][j] = eluf(fmaf(aT[wv][j], ssd0[f], ssd0[64 + f]));
  }
  __syncthreads();
  int m16 = lane & 15, halfk = (lane >> 4) << 1, rof = (lane >> 4) << 3;
  v8f acc[8];
#pragma unroll
  for (int ct = 0; ct < 8; ++ct) acc[ct] = {0.f, 0.f, 0.f, 0.f, 0.f, 0.f, 0.f, 0.f};
#pragma unroll
  for (int s = 0; s < 16; ++s) {
    int ko = 4 * s + halfk;
    v2f a = *(const v2f*)&aT[wv][m16 * 64 + ko];
#pragma unroll
    for (int ct = 0; ct < 8; ++ct) {
      v2f b = *(const v2f*)&B[(ct * 16 + m16) * 64 + ko];
      acc[ct] = wmma4(a, b, acc[ct]);
    }
  }
#pragma unroll
  for (int ct = 0; ct < 8; ++ct) {
    int col = ct * 16 + m16;
    float bia = d1b[col];
    float s_ = 0.f, q_ = 0.f;
#pragma unroll
    for (int i = 0; i < 8; ++i) {
      float v = acc[ct][i] + bia;
      e1[(size_t)(rowBase + i + rof) * 128 + col] = v;
      s_ += v; q_ += v * v;
    }
    atomicAdd(&sSt[col], s_);
    atomicAdd(&sSt[128 + col], q_);
  }
  __syncthreads();
  atomicAdd(&part[(blockIdx.x & 63) * 256 + tid], sSt[tid]);
}

// ---------------- K11: elu(bnd1(e1)) @ d2_w^T (WMMA) -> input_hat ----------------
__global__ __launch_bounds__(256) void k_d2(const float* __restrict__ e1,
                                            const float* __restrict__ ssd1,
                                            const float* __restrict__ d2w,
                                            const float* __restrict__ d2b,
                                            float* __restrict__ outIH) {
  __shared__ __attribute__((aligned(16))) float B[48 * 128];   // pad 45->48 out cols
  __shared__ __attribute__((aligned(16))) float aT[8][16 * 128];
  int tid = threadIdx.x, lane = tid & 31, wv = tid >> 5;
  int rowBase = (blockIdx.x * 8 + wv) * 16;
  lds_cp128(B, d2w, 5760, tid, 256);                           // 45*128 valid floats
  for (int j = 5760 + tid; j < 6144; j += 256) B[j] = 0.f;     // zero pad rows
  lds_cp128(aT[wv], e1 + (size_t)rowBase * 128, 2048, lane, 32);
  lds_cp_wait();
  __syncthreads();
  for (int j = lane; j < 2048; j += 32) {
    int f = j & 127;
    aT[wv][j] = eluf(fmaf(aT[wv][j], ssd1[f], ssd1[128 + f]));
  }
  __syncthreads();
  int m16 = lane & 15, halfk = (lane >> 4) << 1, rof = (lane >> 4) << 3;
  v8f acc[3];
#pragma unroll
  for (int ct = 0; ct < 3; ++ct) acc[ct] = {0.f, 0.f, 0.f, 0.f, 0.f, 0.f, 0.f, 0.f};
#pragma unroll
  for (int s = 0; s < 32; ++s) {
    int ko = 4 * s + halfk;
    v2f a = *(const v2f*)&aT[wv][m16 * 128 + ko];
#pragma unroll
    for (int ct = 0; ct < 3; ++ct) {
      v2f b = *(const v2f*)&B[(ct * 16 + m16) * 128 + ko];
      acc[ct] = wmma4(a, b, acc[ct]);
    }
  }
#pragma unroll
  for (int ct = 0; ct < 3; ++ct) {
    int col = ct * 16 + m16;
    if (col < 45) {
      float bia = d2b[col];
#pragma unroll
      for (int i = 0; i < 8; ++i)
        outIH[(size_t)(rowBase + i + rof) * 45 + col] = acc[ct][i] + bia;
    }
  }
}

// ---------------- host ----------------
extern "C" void kernel_launch(void* const* d_in, const int* in_sizes, int n_in,
                              void* d_out, int out_size, void* d_ws, size_t ws_size,
                              hipStream_t stream) {
  const float* obs     = (const float*)d_in[0];
  const float* enc_w   = (const float*)d_in[1];
  const float* enc_b   = (const float*)d_in[2];
  const float* bn0_g   = (const float*)d_in[3];
  const float* bn0_b   = (const float*)d_in[4];
  const float* conv1_w = (const float*)d_in[5];
  const float* conv1_b = (const float*)d_in[6];
  const float* bn1_g   = (const float*)d_in[7];
  const float* bn1_b   = (const float*)d_in[8];
  const float* conv2_w = (const float*)d_in[9];
  const float* conv2_b = (const float*)d_in[10];
  const float* bn2_g   = (const float*)d_in[11];
  const float* bn2_b   = (const float*)d_in[12];
  const float* lin_w   = (const float*)d_in[13];
  const float* lin_b   = (const float*)d_in[14];
  const float* vel_w   = (const float*)d_in[15];
  const float* vel_b   = (const float*)d_in[16];
  const float* codebook= (const float*)d_in[17];
  const float* d0_w    = (const float*)d_in[18];
  const float* d0_b    = (const float*)d_in[19];
  const float* bnd0_g  = (const float*)d_in[20];
  const float* bnd0_b  = (const float*)d_in[21];
  const float* d1_w    = (const float*)d_in[22];
  const float* d1_b    = (const float*)d_in[23];
  const float* bnd1_g  = (const float*)d_in[24];
  const float* bnd1_b  = (const float*)d_in[25];
  const float* d2_w    = (const float*)d_in[26];
  const float* d2_b    = (const float*)d_in[27];
  float* ws  = (float*)d_ws;
  float* out = (float*)d_out;

  k_zero<<<(PART_TOTAL + 255) / 256, 256, 0, stream>>>(ws + WS_P0, PART_TOTAL);
  k_enc<<<NT / (16 * 8), 256, 0, stream>>>(obs, enc_w, enc_b, ws + WS_H0, ws + WS_P0);
  k_bnfin<<<1, 128, 0, stream>>>(ws + WS_P0, bn0_g, bn0_b, ws + WS_S0, 48, 1.f / (float)NT);
  k_conv1<<<N_ROWS / 8, 256, 0, stream>>>(ws + WS_H0, ws + WS_S0, conv1_w, conv1_b,
                                          ws + WS_C1, ws + WS_P1);
  k_bnfin<<<1, 128, 0, stream>>>(ws + WS_P1, bn1_g, bn1_b, ws + WS_S1, 32,
                                 1.f / (float)(N_ROWS * 4));
  k_conv2<<<N_ROWS / 16, 256, 0, stream>>>(ws + WS_C1, ws + WS_S1, conv2_w, conv2_b,
                                           ws + WS_C2, ws + WS_P2);
  k_bnfin<<<1, 128, 0, stream>>>(ws + WS_P2, bn2_g, bn2_b, ws + WS_S2, 16,
                                 1.f / (float)(N_ROWS * 3));
  k_head<<<N_ROWS / (16 * 8), 256, 0, stream>>>(ws + WS_C2, ws + WS_S2, lin_w, lin_b,
                                                vel_w, vel_b, codebook, d0_w, d0_b,
                                                ws + WS_E0, ws + WS_PD0, out + O_Q,
                                                out + O_Z, out + O_V);
  k_bnfin<<<1, 128, 0, stream>>>(ws + WS_PD0, bnd0_g, bnd0_b, ws + WS_SD0, 64,
                                 1.f / (float)N_ROWS);
  k_d1<<<N_ROWS / (16 * 8), 256, 0, stream>>>(ws + WS_E0, ws + WS_SD0, d1_w, d1_b,
                                              ws + WS_E1, ws + WS_PD1);
  k_bnfin<<<1, 128, 0, stream>>>(ws + WS_PD1, bnd1_g, bnd1_b, ws + WS_SD1, 128,
                                 1.f / (float)N_ROWS);
  k_d2<<<N_ROWS / (16 * 8), 256, 0, stream>>>(ws + WS_E1, ws + WS_SD1, d2_w, d2_b,
                                              out + O_IH);
}